// Half_Graph_51539607552473
// MI455X (gfx1250) — compile-verified
//
#include <hip/hip_runtime.h>
#include <math.h>

#define HH 96
#define WW 96
#define HW 9216          // HH*WW
#define NB 8
#define NPIX 73728       // NB*HW
#define EPS 1e-5f

typedef __attribute__((ext_vector_type(2))) float v2f;
typedef __attribute__((ext_vector_type(8))) float v8f;

__device__ __forceinline__ float sigm(float x) { return 1.f / (1.f + expf(-x)); }

// ---------------------------------------------------------------------------
// datt: grouped 1x1 conv (2 groups of 10) + sigmoid -> att0/att1 (d_out)
// ---------------------------------------------------------------------------
__global__ void k_datt(const float* __restrict__ xh,
                       const float* __restrict__ attw,
                       const float* __restrict__ attb,
                       float* __restrict__ att0, float* __restrict__ att1) {
  int pix = blockIdx.x * blockDim.x + threadIdx.x;
  if (pix >= NPIX) return;
  int b = pix / HW, hw = pix - b * HW;
  const float* xb = xh + (size_t)b * 20 * HW + hw;
  float s0 = attb[0], s1 = attb[1];
#pragma unroll
  for (int c = 0; c < 10; ++c) {
    s0 += attw[c] * xb[(size_t)c * HW];
    s1 += attw[10 + c] * xb[(size_t)(10 + c) * HW];
  }
  att0[pix] = sigm(s0);
  att1[pix] = sigm(s1);
}

// ---------------------------------------------------------------------------
// NCHW -> planar [chan][NPIX] for xh (20 channels)
// ---------------------------------------------------------------------------
__global__ void k_planar_xh(const float* __restrict__ xh, float* __restrict__ dst) {
  int idx = blockIdx.x * blockDim.x + threadIdx.x;
  if (idx >= 20 * NPIX) return;
  int c = idx / NPIX, pix = idx - c * NPIX;
  int b = pix / HW, hw = pix - b * HW;
  dst[idx] = xh[((size_t)b * 20 + c) * HW + hw];
}

// ---------------------------------------------------------------------------
// composition pre: catt sigmoid + group-sum, emit 20 planar act planes
// act[0..9]=xh_half, act[10..19]=psum*att
// ---------------------------------------------------------------------------
__global__ void k_comp_pre(const float* __restrict__ xp, int xpoff, int nparts,
                           const float* __restrict__ xhp, int half,
                           const float* __restrict__ cattw,
                           const float* __restrict__ cattb,
                           float* __restrict__ act) {
  int pix = blockIdx.x * blockDim.x + threadIdx.x;
  if (pix >= NPIX) return;
  int b = pix / HW, hw = pix - b * HW;
  const float* pb = xp + ((size_t)b * 60 + xpoff) * HW + hw;
  float satt = cattb[0];
  float psum[10];
#pragma unroll
  for (int c = 0; c < 10; ++c) psum[c] = 0.f;
  for (int g = 0; g < nparts; ++g) {
#pragma unroll
    for (int c = 0; c < 10; ++c) {
      float v = pb[(size_t)(g * 10 + c) * HW];
      satt += cattw[g * 10 + c] * v;
      psum[c] += v;
    }
  }
  float a = sigm(satt);
#pragma unroll
  for (int c = 0; c < 10; ++c) {
    act[(size_t)c * NPIX + pix] = xhp[(size_t)(half + c) * NPIX + pix];
    act[(size_t)(10 + c) * NPIX + pix] = psum[c] * a;
  }
}

// ---------------------------------------------------------------------------
// WMMA GEMM: D[pix, n] = sum_k A[pix, k] * B[k, n]
//   act: planar [Ktot][NPIX], wgt: [N][Ktot] (natural OIK layout), out: [N][NPIX]
// One wave (32 threads) computes a 16(pix) x 16(outch) tile with
// V_WMMA_F32_16X16X4_F32, looping K in steps of 4. Ktot % 4 == 0 required.
// ---------------------------------------------------------------------------
__global__ void k_gemm_wmma(const float* __restrict__ act,
                            const float* __restrict__ wgt,
                            float* __restrict__ out, int Ktot, int N) {
  const int lane = threadIdx.x;         // blockDim == 32, full wave, EXEC all 1s
  const int hi = lane >> 4;             // 0: lanes 0-15, 1: lanes 16-31
  const int lo = lane & 15;
  const int pix0 = blockIdx.x << 4;
  const int n = (blockIdx.y << 4) + lo;
  const int nn = (n < N) ? n : 0;
  const float msk = (n < N) ? 1.f : 0.f;
  const float* ap = act + pix0 + lo;    // A: M = lo
  const float* wp = wgt + (size_t)nn * Ktot;
  v8f acc = {0.f, 0.f, 0.f, 0.f, 0.f, 0.f, 0.f, 0.f};
  for (int k = 0; k < Ktot; k += 4) {
    const int ka = k + (hi << 1);       // VGPR0: K = k (+2 for hi lanes)
    v2f a, bm;
    a.x = ap[(size_t)ka * NPIX];
    a.y = ap[(size_t)(ka + 1) * NPIX];  // VGPR1: K = k+1 (+2 for hi lanes)
    bm.x = wp[ka] * msk;                // B mirrors A's K striping, N = lo
    bm.y = wp[ka + 1] * msk;
    acc = __builtin_amdgcn_wmma_f32_16x16x4_f32(
        /*neg_a=*/false, a, /*neg_b=*/false, bm,
        /*c_mod=*/(short)0, acc, /*reuse_a=*/false, /*reuse_b=*/false);
  }
  if (n < N) {
    // D VGPR i: M = i + 8*hi, N = lo
    float* op = out + (size_t)n * NPIX + pix0 + (hi << 3);
#pragma unroll
    for (int i = 0; i < 8; ++i) op[i] = acc[i];
  }
}

// ---------------------------------------------------------------------------
// BN stats: one block per channel, deterministic tree reduction
// stats[2c] = mean, stats[2c+1] = var (biased)
// ---------------------------------------------------------------------------
__global__ void k_bn_stats(const float* __restrict__ x, float* __restrict__ stats) {
  __shared__ float ssum[256];
  __shared__ float ssq[256];
  int c = blockIdx.x;
  const float* p = x + (size_t)c * NPIX;
  float s = 0.f, q = 0.f;
  for (int i = threadIdx.x; i < NPIX; i += 256) {
    float v = p[i];
    s += v;
    q += v * v;
  }
  ssum[threadIdx.x] = s;
  ssq[threadIdx.x] = q;
  __syncthreads();
  for (int off = 128; off > 0; off >>= 1) {
    if (threadIdx.x < off) {
      ssum[threadIdx.x] += ssum[threadIdx.x + off];
      ssq[threadIdx.x] += ssq[threadIdx.x + off];
    }
    __syncthreads();
  }
  if (threadIdx.x == 0) {
    float mean = ssum[0] * (1.f / (float)NPIX);
    float var = ssq[0] * (1.f / (float)NPIX) - mean * mean;
    stats[2 * c] = mean;
    stats[2 * c + 1] = var;
  }
}

// BN + ReLU, planar -> planar (in-place safe)
__global__ void k_bn_apply(const float* __restrict__ pre, float* __restrict__ out,
                           const float* __restrict__ stats,
                           const float* __restrict__ g, const float* __restrict__ bb,
                           int total) {
  int idx = blockIdx.x * blockDim.x + threadIdx.x;
  if (idx >= total) return;
  int c = idx / NPIX;
  float mean = stats[2 * c], var = stats[2 * c + 1];
  float y = (pre[idx] - mean) * rsqrtf(var + EPS) * g[c] + bb[c];
  out[idx] = y > 0.f ? y : 0.f;
}

// BN + ReLU then * (1 - attA) * attB  (part_dependency tail), 10 channels
__global__ void k_bn_apply_pd(const float* __restrict__ pre, float* __restrict__ out,
                              const float* __restrict__ stats,
                              const float* __restrict__ g, const float* __restrict__ bb,
                              const float* __restrict__ attA,
                              const float* __restrict__ attB) {
  int idx = blockIdx.x * blockDim.x + threadIdx.x;
  if (idx >= 10 * NPIX) return;
  int c = idx / NPIX, pix = idx - c * NPIX;
  float mean = stats[2 * c], var = stats[2 * c + 1];
  float y = (pre[idx] - mean) * rsqrtf(var + EPS) * g[c] + bb[c];
  y = y > 0.f ? y : 0.f;
  out[idx] = y * (1.f - attA[pix]) * attB[pix];
}

// BN + ReLU + residual, write natural NCHW into d_out slice (conv_update tail)
__global__ void k_bn_apply_update(const float* __restrict__ pre,
                                  const float* __restrict__ stats,
                                  const float* __restrict__ g,
                                  const float* __restrict__ bb,
                                  const float* __restrict__ xhp, int half,
                                  float* __restrict__ out_nat) {
  int idx = blockIdx.x * blockDim.x + threadIdx.x;
  if (idx >= 10 * NPIX) return;
  int c = idx / NPIX, pix = idx - c * NPIX;
  int b = pix / HW, hw = pix - b * HW;
  float mean = stats[2 * c], var = stats[2 * c + 1];
  float y = (pre[idx] - mean) * rsqrtf(var + EPS) * g[c] + bb[c];
  y = y > 0.f ? y : 0.f;
  out_nat[((size_t)b * 10 + c) * HW + hw] = xhp[(size_t)(half + c) * NPIX + pix] + y;
}

// ---------------------------------------------------------------------------
// 3x3 SAME offset conv, 20 -> 27, planar in/out; chanoff rotates input planes
// ---------------------------------------------------------------------------
__global__ void k_off_conv(const float* __restrict__ xin, int chanoff,
                           const float* __restrict__ w,
                           const float* __restrict__ bias,
                           float* __restrict__ om) {
  int t = blockIdx.x * blockDim.x + threadIdx.x;
  if (t >= 27 * NPIX) return;
  int oc = t / NPIX, pix = t - oc * NPIX;
  int b = pix / HW, hw = pix - b * HW;
  int y = hw / WW, x = hw - y * WW;
  const float* xb = xin + (size_t)b * HW;
  float s = bias[oc];
#pragma unroll
  for (int ky = -1; ky <= 1; ++ky) {
    int yy = y + ky;
    if (yy < 0 || yy >= HH) continue;
#pragma unroll
    for (int kx = -1; kx <= 1; ++kx) {
      int xx = x + kx;
      if (xx < 0 || xx >= WW) continue;
      int tap = (ky + 1) * 3 + (kx + 1);
      const float* wr = w + oc * 180 + tap;  // w[oc][c][tap], OIHW
      int off = yy * WW + xx;
#pragma unroll
      for (int c = 0; c < 20; ++c) {
        int plane = c + chanoff;
        if (plane >= 20) plane -= 20;
        s += wr[c * 9] * xb[(size_t)plane * NPIX + off];
      }
    }
  }
  om[(size_t)oc * NPIX + pix] = s;
}

// ---------------------------------------------------------------------------
// modulated deformable sampling: per (pix, tap k) bilinear-sample 20 channels,
// multiply sigmoid mask, write planar v[(c*9+k)][pix]
// om planes: [2k]=dy, [2k+1]=dx, [18+k]=mask logits
// ---------------------------------------------------------------------------
__global__ void k_sample(const float* __restrict__ xin, int chanoff,
                         const float* __restrict__ om,
                         float* __restrict__ vbuf) {
  int t = blockIdx.x * blockDim.x + threadIdx.x;
  if (t >= 9 * NPIX) return;
  int k = t / NPIX, pix = t - k * NPIX;
  int b = pix / HW, hw = pix - b * HW;
  int y = hw / WW, x = hw - y * WW;
  float offy = om[(size_t)(2 * k) * NPIX + pix];
  float offx = om[(size_t)(2 * k + 1) * NPIX + pix];
  float m = sigm(om[(size_t)(18 + k) * NPIX + pix]);
  float py = (float)(y + (k / 3) - 1) + offy;
  float px = (float)(x + (k % 3) - 1) + offx;
  float fy = floorf(py), fx = floorf(px);
  float dy = py - fy, dx = px - fx;
  int y0 = (int)fy, x0 = (int)fx;
  float wgt4[4];
  int idx4[4];
#pragma unroll
  for (int cy = 0; cy < 2; ++cy) {
    int yy = y0 + cy;
    float wy = cy ? dy : 1.f - dy;
    int yc = yy < 0 ? 0 : (yy > HH - 1 ? HH - 1 : yy);
    float vy = (yy >= 0 && yy < HH) ? 1.f : 0.f;
#pragma unroll
    for (int cx = 0; cx < 2; ++cx) {
      int xx = x0 + cx;
      float wx = cx ? dx : 1.f - dx;
      int xc = xx < 0 ? 0 : (xx > WW - 1 ? WW - 1 : xx);
      float vx = (xx >= 0 && xx < WW) ? 1.f : 0.f;
      wgt4[cy * 2 + cx] = wy * wx * vy * vx * m;
      idx4[cy * 2 + cx] = yc * WW + xc;
    }
  }
  const float* xb = xin + (size_t)b * HW;
#pragma unroll
  for (int c = 0; c < 20; ++c) {
    int plane = c + chanoff;
    if (plane >= 20) plane -= 20;
    const float* xc_ = xb + (size_t)plane * NPIX;
    float v = wgt4[0] * xc_[idx4[0]] + wgt4[1] * xc_[idx4[1]] +
              wgt4[2] * xc_[idx4[2]] + wgt4[3] * xc_[idx4[3]];
    vbuf[(size_t)(c * 9 + k) * NPIX + pix] = v;
  }
}

// ---------------------------------------------------------------------------
// decomposition pre: att sigmoid (-> d_out att_u/att_l) + 20 planar act planes
// act[0..9] = xf*att, act[10..19] = xh_half
// ---------------------------------------------------------------------------
__global__ void k_dec_pre(const float* __restrict__ xf,
                          const float* __restrict__ xhp, int half,
                          const float* __restrict__ attw,
                          const float* __restrict__ attb,
                          float* __restrict__ act, float* __restrict__ attout) {
  int pix = blockIdx.x * blockDim.x + threadIdx.x;
  if (pix >= NPIX) return;
  int b = pix / HW, hw = pix - b * HW;
  const float* fb = xf + (size_t)b * 10 * HW + hw;
  float s = attb[0];
  float fv[10], hv[10];
#pragma unroll
  for (int c = 0; c < 10; ++c) {
    fv[c] = fb[(size_t)c * HW];
    hv[c] = xhp[(size_t)(half + c) * NPIX + pix];
    s += attw[c] * fv[c] + attw[10 + c] * hv[c];
  }
  float a = sigm(s);
  attout[pix] = a;
#pragma unroll
  for (int c = 0; c < 10; ++c) {
    act[(size_t)c * NPIX + pix] = fv[c] * a;
    act[(size_t)(10 + c) * NPIX + pix] = hv[c];
  }
}

// conv_update input: 40 planar planes = [xh_half, m1, m2, m3]
__global__ void k_upd_pre(const float* __restrict__ xhp, int half,
                          const float* __restrict__ m1,
                          const float* __restrict__ m2,
                          const float* __restrict__ m3,
                          float* __restrict__ act) {
  int idx = blockIdx.x * blockDim.x + threadIdx.x;
  if (idx >= 40 * NPIX) return;
  int p = idx / NPIX, pix = idx - p * NPIX;
  float v;
  if (p < 10) v = xhp[(size_t)(half + p) * NPIX + pix];
  else if (p < 20) v = m1[(size_t)(p - 10) * NPIX + pix];
  else if (p < 30) v = m2[(size_t)(p - 20) * NPIX + pix];
  else v = m3[(size_t)(p - 30) * NPIX + pix];
  act[idx] = v;
}

// ---------------------------------------------------------------------------
// Host side
// ---------------------------------------------------------------------------
struct BranchP {
  int half, xpoff, nparts;
  const float *catt_b, *catt_w, *ch_b, *ch_g, *ch_w;
  const float *d1_b, *d1_g, *d1_off_b, *d1_off_w, *d1_w;
  const float *d2_b, *d2_g, *d2_off_b, *d2_off_w, *d2_w;
  const float *datt_b, *datt_w, *fh_b, *fh_g, *fh_w;
  const float *up_b, *up_g, *up_w;
  const float *attA, *attB;  // result scaled by (1-attA)*attB
  float *att_out, *xh_out;   // att_u/att_l slice, xh_u/xh_l slice (NCHW)
};

static inline void run_gemm(const float* act, const float* wgt, float* out,
                            int K, int N, hipStream_t s) {
  dim3 grid(NPIX / 16, (N + 15) / 16);
  k_gemm_wmma<<<grid, dim3(32), 0, s>>>(act, wgt, out, K, N);
}

#define TPB 256
#define GB(n) (((n) + TPB - 1) / TPB)

static void run_branch(const BranchP& p, const float* xf, const float* xp,
                       float* xhp, float* vbuf, float* om, float* d1b,
                       float* mcomp, float* mpd, float* mdec, float* pre,
                       float* stats, hipStream_t s) {
  // composition
  k_comp_pre<<<GB(NPIX), TPB, 0, s>>>(xp, p.xpoff, p.nparts, xhp, p.half,
                                      p.catt_w, p.catt_b, vbuf);
  run_gemm(vbuf, p.ch_w, pre, 20, 10, s);
  k_bn_stats<<<10, 256, 0, s>>>(pre, stats);
  k_bn_apply<<<GB(10 * NPIX), TPB, 0, s>>>(pre, mcomp, stats, p.ch_g, p.ch_b,
                                           10 * NPIX);
  // part_dependency: dfconv d1 (input = rotated xh planes)
  k_off_conv<<<GB(27 * NPIX), TPB, 0, s>>>(xhp, p.half, p.d1_off_w, p.d1_off_b, om);
  k_sample<<<GB(9 * NPIX), TPB, 0, s>>>(xhp, p.half, om, vbuf);
  run_gemm(vbuf, p.d1_w, d1b, 180, 20, s);
  k_bn_stats<<<20, 256, 0, s>>>(d1b, stats);
  k_bn_apply<<<GB(20 * NPIX), TPB, 0, s>>>(d1b, d1b, stats, p.d1_g, p.d1_b,
                                           20 * NPIX);
  // dfconv d2
  k_off_conv<<<GB(27 * NPIX), TPB, 0, s>>>(d1b, 0, p.d2_off_w, p.d2_off_b, om);
  k_sample<<<GB(9 * NPIX), TPB, 0, s>>>(d1b, 0, om, vbuf);
  run_gemm(vbuf, p.d2_w, pre, 180, 10, s);
  k_bn_stats<<<10, 256, 0, s>>>(pre, stats);
  k_bn_apply_pd<<<GB(10 * NPIX), TPB, 0, s>>>(pre, mpd, stats, p.d2_g, p.d2_b,
                                              p.attA, p.attB);
  // decomposition
  k_dec_pre<<<GB(NPIX), TPB, 0, s>>>(xf, xhp, p.half, p.datt_w, p.datt_b, vbuf,
                                     p.att_out);
  run_gemm(vbuf, p.fh_w, pre, 20, 10, s);
  k_bn_stats<<<10, 256, 0, s>>>(pre, stats);
  k_bn_apply<<<GB(10 * NPIX), TPB, 0, s>>>(pre, mdec, stats, p.fh_g, p.fh_b,
                                           10 * NPIX);
  // conv_update + residual
  k_upd_pre<<<GB(40 * NPIX), TPB, 0, s>>>(xhp, p.half, mcomp, mpd, mdec, vbuf);
  run_gemm(vbuf, p.up_w, pre, 40, 10, s);
  k_bn_stats<<<10, 256, 0, s>>>(pre, stats);
  k_bn_apply_update<<<GB(10 * NPIX), TPB, 0, s>>>(pre, stats, p.up_g, p.up_b,
                                                  xhp, p.half, p.xh_out);
}

extern "C" void kernel_launch(void* const* d_in, const int* in_sizes, int n_in,
                              void* d_out, int out_size, void* d_ws, size_t ws_size,
                              hipStream_t stream) {
  (void)in_sizes; (void)out_size; (void)ws_size;
  if (n_in < 52) return;
#define P(i) ((const float*)d_in[(i)])
  // top-level insertion order: h_fea(0, unused), xh(1), xf(2), xp(3),
  // then params leaves in recursive-alphabetical (pytree) order.
  const float* xh = P(1);
  const float* xf = P(2);
  const float* xp = P(3);
  const float* att_b = P(4);
  const float* att_w = P(5);
  // comp_l: catt_b, catt_w, ch_b, ch_g, ch_w  (6..10)
  // comp_u: (11..15)
  // dec_l:  att_b, att_w, fh_b, fh_g, fh_w    (16..20)
  // dec_u:  (21..25)
  // pd_h:   d1_b,d1_g,d1_off_b,d1_off_w,d1_w,d2_b,d2_g,d2_off_b,d2_off_w,d2_w (26..35)
  // pd_l:   (36..45)
  // up_l:   b, g, w (46..48);  up_u: (49..51)

  float* out = (float*)d_out;
  float* o_xhu = out;
  float* o_xhl = out + (size_t)10 * NPIX;
  float* o_attu = out + (size_t)20 * NPIX;
  float* o_attl = out + (size_t)21 * NPIX;
  float* o_att0 = out + (size_t)22 * NPIX;
  float* o_att1 = out + (size_t)23 * NPIX;

  float* w0 = (float*)d_ws;
  float* xhp   = w0;                           // 20 planes
  float* vbuf  = xhp  + (size_t)20 * NPIX;     // 180 planes (also act20/act40)
  float* om    = vbuf + (size_t)180 * NPIX;    // 27 planes
  float* d1b   = om   + (size_t)27 * NPIX;     // 20 planes
  float* mcomp = d1b  + (size_t)20 * NPIX;     // 10 planes
  float* mpd   = mcomp+ (size_t)10 * NPIX;     // 10 planes
  float* mdec  = mpd  + (size_t)10 * NPIX;     // 10 planes
  float* pre   = mdec + (size_t)10 * NPIX;     // 10 planes
  float* stats = pre  + (size_t)10 * NPIX;     // 64 floats

  k_datt<<<GB(NPIX), TPB, 0, stream>>>(xh, att_w, att_b, o_att0, o_att1);
  k_planar_xh<<<GB(20 * NPIX), TPB, 0, stream>>>(xh, xhp);

  BranchP up;
  up.half = 0; up.xpoff = 0; up.nparts = 4;
  up.catt_b = P(11); up.catt_w = P(12); up.ch_b = P(13); up.ch_g = P(14); up.ch_w = P(15);
  up.d1_b = P(26); up.d1_g = P(27); up.d1_off_b = P(28); up.d1_off_w = P(29); up.d1_w = P(30);
  up.d2_b = P(31); up.d2_g = P(32); up.d2_off_b = P(33); up.d2_off_w = P(34); up.d2_w = P(35);
  up.datt_b = P(21); up.datt_w = P(22); up.fh_b = P(23); up.fh_g = P(24); up.fh_w = P(25);
  up.up_b = P(49); up.up_g = P(50); up.up_w = P(51);
  up.attA = o_att1; up.attB = o_att0;          // xlh = (1-att1)*x*att0
  up.att_out = o_attu; up.xh_out = o_xhu;
  run_branch(up, xf, xp, xhp, vbuf, om, d1b, mcomp, mpd, mdec, pre, stats, stream);

  BranchP lo;
  lo.half = 10; lo.xpoff = 40; lo.nparts = 2;
  lo.catt_b = P(6); lo.catt_w = P(7); lo.ch_b = P(8); lo.ch_g = P(9); lo.ch_w = P(10);
  lo.d1_b = P(36); lo.d1_g = P(37); lo.d1_off_b = P(38); lo.d1_off_w = P(39); lo.d1_w = P(40);
  lo.d2_b = P(41); lo.d2_g = P(42); lo.d2_off_b = P(43); lo.d2_off_w = P(44); lo.d2_w = P(45);
  lo.datt_b = P(16); lo.datt_w = P(17); lo.fh_b = P(18); lo.fh_g = P(19); lo.fh_w = P(20);
  lo.up_b = P(46); lo.up_g = P(47); lo.up_w = P(48);
  lo.attA = o_att0; lo.attB = o_att1;          // xuh = (1-att0)*x*att1
  lo.att_out = o_attl; lo.xh_out = o_xhl;
  run_branch(lo, xf, xp, xhp, vbuf, om, d1b, mcomp, mpd, mdec, pre, stats, stream);
#undef P
}